// BasicConvolutionBlock_2001454760193
// MI455X (gfx1250) — compile-verified
//
#include <hip/hip_runtime.h>

typedef float v2f __attribute__((ext_vector_type(2)));
typedef float v8f __attribute__((ext_vector_type(8)));

#define NPOINTS 300000
#define NPAIRS  100000
#define KOFF    27
#define CIN     32
#define COUT    64
#define NGROUPS 8
#define EPSV    1e-5f
#define SLOPE   0.01f

// ---------------------------------------------------------------------------
// Kernel 1: zero the accumulation output + the 16 group-stat accumulators
// ---------------------------------------------------------------------------
__global__ __launch_bounds__(256) void zero_kernel(float4* __restrict__ out4,
                                                   float* __restrict__ stats,
                                                   int n4) {
  int stride = gridDim.x * blockDim.x;
  for (int i = blockIdx.x * blockDim.x + threadIdx.x; i < n4; i += stride) {
    out4[i] = make_float4(0.f, 0.f, 0.f, 0.f);
  }
  if (blockIdx.x == 0 && threadIdx.x < 2 * NGROUPS) stats[threadIdx.x] = 0.f;
}

// ---------------------------------------------------------------------------
// Kernel 2: rulebook gather -> WMMA f32 16x16x4 matmul -> atomic scatter-add
// One wave32 handles 16 pairs of one kernel offset. blockIdx.x = offset k,
// blockIdx.y tiles the pair dimension (8 waves per 256-thread block).
// ---------------------------------------------------------------------------
__global__ __launch_bounds__(256)
void conv_scatter(const float* __restrict__ feats,
                  const float* __restrict__ weight,
                  const int* __restrict__ in_idx,
                  const int* __restrict__ out_idx,
                  float* __restrict__ out) {
  __shared__ float wlds[CIN * COUT];  // 8 KB weight slice for this offset

  const int kk  = blockIdx.x;
  const int tid = threadIdx.x;

  // Stage this offset's 32x64 weight slice into LDS (float4 wide).
  const float4* wsrc = (const float4*)(weight + kk * CIN * COUT);
  for (int i = tid; i < (CIN * COUT) / 4; i += 256) {
    ((float4*)wlds)[i] = wsrc[i];
  }
  __syncthreads();

  const int wave = tid >> 5;
  const int lane = tid & 31;
  const int tile = blockIdx.y * 8 + wave;
  const int p0   = tile * 16;
  if (p0 >= NPAIRS) return;  // wave-uniform: EXEC stays all-1s for WMMA

  const int m  = lane & 15;          // matrix row (pair within tile)
  const int kb = (lane >> 4) << 1;   // K sub-column: 0 for lanes 0-15, 2 for 16-31

  const int inrow  = in_idx [kk * NPAIRS + p0 + m];
  const int outrow = out_idx[kk * NPAIRS + p0 + m];

  // A-matrix: 8 K-steps of 16x4 f32. Lane layout: VGPR0=K(kb), VGPR1=K(kb+1).
  const float* frow = feats + inrow * CIN;
  v2f a[8];
#pragma unroll
  for (int ks = 0; ks < 8; ++ks) {
    a[ks] = *(const v2f*)(frow + ks * 4 + kb);  // 8-byte aligned pair load
  }

  // Broadcast the 16 output rows per C/D layout: VGPR v <-> row (v + 8*hi).
  const int mbase = (lane >> 4) << 3;
  int orow[8];
#pragma unroll
  for (int v = 0; v < 8; ++v) orow[v] = __shfl(outrow, mbase + v, 32);

#pragma unroll
  for (int nt = 0; nt < 4; ++nt) {
    const int n = nt * 16 + m;  // output column for this lane
    v8f acc = {};
#pragma unroll
    for (int ks = 0; ks < 8; ++ks) {
      v2f b;
      b.x = wlds[(ks * 4 + kb)     * COUT + n];
      b.y = wlds[(ks * 4 + kb + 1) * COUT + n];
      acc = __builtin_amdgcn_wmma_f32_16x16x4_f32(
          /*neg_a=*/false, a[ks], /*neg_b=*/false, b,
          /*c_mod=*/(short)0, acc, /*reuse_a=*/false, /*reuse_b=*/false);
    }
#pragma unroll
    for (int v = 0; v < 8; ++v) {
      atomicAdd(&out[orow[v] * COUT + n], acc[v]);
    }
  }
}

// ---------------------------------------------------------------------------
// Kernel 3: per-group sum / sum-of-squares. Stride is a multiple of 64, so
// each thread's channel (i % 64) is fixed -> private accumulation, then LDS
// atomics, then 16 global atomics per block.
// ---------------------------------------------------------------------------
__global__ __launch_bounds__(256)
void group_stats(const float* __restrict__ out, float* __restrict__ stats, int n) {
  __shared__ float s[2 * NGROUPS];
  const int tid = threadIdx.x;
  if (tid < 2 * NGROUPS) s[tid] = 0.f;
  __syncthreads();

  float sum = 0.f, sq = 0.f;
  const int stride = gridDim.x * blockDim.x;  // multiple of 64
  for (int i = blockIdx.x * blockDim.x + tid; i < n; i += stride) {
    float x = out[i];
    sum += x;
    sq  += x * x;
  }
  const int g = (tid & 63) >> 3;  // channel = tid % 64 (invariant), group = c/8
  atomicAdd(&s[g * 2], sum);
  atomicAdd(&s[g * 2 + 1], sq);
  __syncthreads();
  if (tid < 2 * NGROUPS) atomicAdd(&stats[tid], s[tid]);
}

// ---------------------------------------------------------------------------
// Kernel 4: GroupNorm normalize + affine + LeakyReLU, in place.
// ---------------------------------------------------------------------------
__global__ __launch_bounds__(256)
void norm_act(float* __restrict__ out, const float* __restrict__ stats,
              const float* __restrict__ gamma, const float* __restrict__ beta,
              int n) {
  const float invcnt = 1.0f / (float)(NPOINTS * (COUT / NGROUPS));
  const int stride = gridDim.x * blockDim.x;
  for (int i = blockIdx.x * blockDim.x + threadIdx.x; i < n; i += stride) {
    const int c = i & 63;
    const int g = c >> 3;
    const float mean = stats[g * 2] * invcnt;
    const float var  = stats[g * 2 + 1] * invcnt - mean * mean;
    const float rs   = rsqrtf(var + EPSV);
    float x = (out[i] - mean) * rs * gamma[c] + beta[c];
    out[i] = (x >= 0.f) ? x : SLOPE * x;
  }
}

// ---------------------------------------------------------------------------
extern "C" void kernel_launch(void* const* d_in, const int* in_sizes, int n_in,
                              void* d_out, int out_size, void* d_ws, size_t ws_size,
                              hipStream_t stream) {
  const float* feats   = (const float*)d_in[0];
  const float* weight  = (const float*)d_in[1];
  const float* gamma   = (const float*)d_in[2];
  const float* beta    = (const float*)d_in[3];
  const int*   in_idx  = (const int*)d_in[4];
  const int*   out_idx = (const int*)d_in[5];
  float*       out     = (float*)d_out;
  float*       stats   = (float*)d_ws;  // 16 floats: {sum, sumsq} x 8 groups

  const int n = NPOINTS * COUT;

  zero_kernel<<<2048, 256, 0, stream>>>((float4*)out, stats, n / 4);

  // 27 offsets x ceil(6250 tiles / 8 waves-per-block) blocks
  dim3 grid(KOFF, (NPAIRS / 16 + 7) / 8);
  conv_scatter<<<grid, 256, 0, stream>>>(feats, weight, in_idx, out_idx, out);

  group_stats<<<2048, 256, 0, stream>>>(out, stats, n);
  norm_act<<<2048, 256, 0, stream>>>(out, stats, gamma, beta, n);
}